// SparseFactorisationDense_44830868635743
// MI455X (gfx1250) — compile-verified
//
#include <hip/hip_runtime.h>

// ---------------------------------------------------------------------------
// SparseFactorisationDense on MI455X (gfx1250)
//
// out = relu(scaling * (x @ W0 @ W1 @ W2) + bias),  x:[16384,2048] f32
// Each Wf is a butterfly: row r nonzero at cols (r + 128k + f) % 2048, k<16.
//
// Composite Wc = W0@W1@W2 is again 16-nnz/row (shift 3), and is block-
// diagonal under the mod-128 column grouping: 128 independent dense 16x16
// blocks -> exact f32 WMMA (V_WMMA_F32_16X16X4_F32), no wasted MACs.
// Memory bound: ~256 MB traffic / 23.3 TB/s ~ 11 us.
//
// x staging uses the gfx1250 async global->LDS pipe (ASYNCcnt) when the
// builtin is available; falls back to b128 load + ds_store otherwise.
// ---------------------------------------------------------------------------

#define NDIM 2048
#define BATCH 16384

typedef float v2f __attribute__((ext_vector_type(2)));
typedef float v8f __attribute__((ext_vector_type(8)));
typedef int sfd_v4i __attribute__((vector_size(16)));  // int __vector(4)

#if __has_builtin(__builtin_amdgcn_global_load_async_to_lds_b128)
#define SFD_ASYNC_LDS 1
typedef __attribute__((address_space(1))) sfd_v4i* sfd_gptr;
typedef __attribute__((address_space(3))) sfd_v4i* sfd_lptr;
#else
#define SFD_ASYNC_LDS 0
#endif

// ---------------------------------------------------------------------------
// Kernel A: compose the three sparse factors into 128 dense 16x16 blocks.
// M[m][i][j] (m<128, i,j<16):  y[b, m'+128j] += x[b, m+128i] * M[m][i][j]
// where m' = (m+3)&127.
// ---------------------------------------------------------------------------
__global__ void sfd_build_blocks(const float* __restrict__ v0,
                                 const float* __restrict__ v1,
                                 const float* __restrict__ v2,
                                 float* __restrict__ M) {
  const int r = blockIdx.x * blockDim.x + threadIdx.x;
  if (r >= NDIM) return;

  float a0[16], w01[16], wc[16];
#pragma unroll
  for (int k = 0; k < 16; ++k) a0[k] = v0[r * 16 + k];
#pragma unroll
  for (int t = 0; t < 16; ++t) w01[t] = 0.f;

  // Stage 1: W0@W1 row r -> w01[t], t = (k0+k1) mod 16, pattern shift 1.
#pragma unroll
  for (int k0 = 0; k0 < 16; ++k0) {
    const int a = (r + 128 * k0) & (NDIM - 1);
#pragma unroll
    for (int t = 0; t < 16; ++t) {
      const int k1 = (t - k0) & 15;
      w01[t] += a0[k0] * v1[a * 16 + k1];
    }
  }

#pragma unroll
  for (int s = 0; s < 16; ++s) wc[s] = 0.f;

  // Stage 2: (W0@W1)@W2 row r -> wc[s], s = (t+k2) mod 16, pattern shift 3.
#pragma unroll
  for (int t = 0; t < 16; ++t) {
    const int bb = (r + 128 * t + 1) & (NDIM - 1);
#pragma unroll
    for (int s = 0; s < 16; ++s) {
      const int k2 = (s - t) & 15;
      wc[s] += w01[t] * v2[bb * 16 + k2];
    }
  }

  // Scatter into block-matrix layout. r = m + 128*i.
  const int m = r & 127;
  const int i = r >> 7;
  const int carry = (m + 3) >> 7;  // +3 shift may cross the mod-128 boundary
#pragma unroll
  for (int s = 0; s < 16; ++s) {
    const int j = (i + s + carry) & 15;
    M[(m * 16 + i) * 16 + j] = wc[s];
  }
}

// ---------------------------------------------------------------------------
// Kernel B: fused block-diagonal GEMM via f32 WMMA + scale/bias/relu.
// Workgroup: 8 waves. Tile: 16 batch rows x 32 residue classes.
// LDS tile[row][i*32 + dm], row stride 516 (pad 4) to spread banks.
// ---------------------------------------------------------------------------
#define RS 516  // padded LDS row stride in floats (516*4 bytes, 16B aligned)

__global__ __launch_bounds__(256) void sfd_wmma(const float* __restrict__ x,
                                                const float* __restrict__ M,
                                                const float* __restrict__ scaling,
                                                const float* __restrict__ bias,
                                                float* __restrict__ out) {
  __shared__ float tile[16 * RS];  // 33,024 B

  const int b0 = blockIdx.x * 16;  // batch tile
  const int m0 = blockIdx.y * 32;  // residue-class group
  const int tid = threadIdx.x;

  // Stage x[b0:b0+16, m0 + 128*i + dm] (i<16, dm<32) into LDS.
  // Each (row,i) pair's 32 floats are one contiguous 128 B cache line.
  // Uses the CDNA5 async global->LDS DMA path (no VGPR transit) when present.
  {
    const int dm4 = (tid & 7) * 4;  // 0,4,...,28
    const int ri = tid >> 3;        // 0..31
#pragma unroll
    for (int p = 0; p < 8; ++p) {
      const int comb = ri + p * 32;  // 0..255 = (row,i)
      const int row = comb & 15;
      const int i = comb >> 4;
      const float* gsrc =
          x + (size_t)(b0 + row) * NDIM + m0 + 128 * i + dm4;
      float* ldst = &tile[row * RS + i * 32 + dm4];
#if SFD_ASYNC_LDS
      __builtin_amdgcn_global_load_async_to_lds_b128(
          (sfd_gptr)gsrc, (sfd_lptr)ldst,
          /*imm offset=*/0, /*cpol=*/0);
#else
      *reinterpret_cast<float4*>(ldst) =
          *reinterpret_cast<const float4*>(gsrc);
#endif
    }
  }
#if SFD_ASYNC_LDS && __has_builtin(__builtin_amdgcn_s_wait_asynccnt)
  __builtin_amdgcn_s_wait_asynccnt(0);  // this wave's DMA writes landed in LDS
#endif
  __syncthreads();

  const int lane = tid & 31;
  const int wave = tid >> 5;
  const int lrow = lane & 15;  // A-matrix M index / B,C,D N index
  const int h = lane >> 4;     // lane half
  const float sc = scaling[0];

#pragma unroll
  for (int ii = 0; ii < 4; ++ii) {
    const int dm = wave * 4 + ii;
    const int m = m0 + dm;

    v8f acc = {};
#pragma unroll
    for (int q = 0; q < 4; ++q) {
      // A 16x4 f32 layout: A[v][lane] = A[row=lane%16][k = v + 2*(lane/16)]
      const int i0 = 4 * q + 2 * h;
      v2f a, b;
      a.x = tile[lrow * RS + (i0 + 0) * 32 + dm];
      a.y = tile[lrow * RS + (i0 + 1) * 32 + dm];
      // B 4x16 f32 layout (mirror): B[v][lane] = B[k = v + 2*(lane/16)][n=lane%16]
      b.x = M[(m * 16 + i0 + 0) * 16 + lrow];
      b.y = M[(m * 16 + i0 + 1) * 16 + lrow];
      acc = __builtin_amdgcn_wmma_f32_16x16x4_f32(
          false, a, false, b, (short)0, acc, false, false);
    }

    // Epilogue: out[b, m'+128j] = relu(sc*acc + bias).
    // C/D layout: acc[v] holds row (v + 8*(lane/16)), col lane%16.
    const int mp = (m + 3) & 127;
    const int c = mp + 128 * lrow;
    const float bz = bias[c];
#pragma unroll
    for (int v = 0; v < 8; ++v) {
      const int brow = b0 + v + 8 * h;
      const float val = sc * acc[v] + bz;
      out[(size_t)brow * NDIM + c] = fmaxf(val, 0.f);
    }
  }
}

// ---------------------------------------------------------------------------
extern "C" void kernel_launch(void* const* d_in, const int* in_sizes, int n_in,
                              void* d_out, int out_size, void* d_ws,
                              size_t ws_size, hipStream_t stream) {
  (void)in_sizes; (void)n_in; (void)out_size; (void)ws_size;
  const float* x  = (const float*)d_in[0];
  const float* v0 = (const float*)d_in[1];
  const float* v1 = (const float*)d_in[2];
  const float* v2 = (const float*)d_in[3];
  // d_in[4..9] are the rows/cols index arrays; the pattern is closed-form.
  const float* scaling = (const float*)d_in[10];
  const float* bias    = (const float*)d_in[11];
  float* out = (float*)d_out;
  float* M = (float*)d_ws;  // 128*16*16 f32 = 128 KB scratch

  sfd_build_blocks<<<8, 256, 0, stream>>>(v0, v1, v2, M);

  dim3 grid(BATCH / 16, 4);  // 1024 batch tiles x 4 residue groups of 32
  sfd_wmma<<<grid, 256, 0, stream>>>(x, M, scaling, bias, out);
}